// HomographyWarper_75385265979468
// MI455X (gfx1250) — compile-verified
//
#include <hip/hip_runtime.h>
#include <hip/hip_bf16.h>

// Homography warp + bilinear sample (zeros padding, align_corners=True).
// patch_src: [16,3,1024,1024] f32, dst_homo_src: [16,3,3] f32 -> out [16,3,1024,1024] f32.
//
// Per wave: projective transform of 32 pixels done with two
// V_WMMA_F32_16X16X4_F32 ops (A = homography rows 0..2 duplicated into rows
// 8..10; B columns = (gx, gy, 1, 0)).  D layout puts row r in VGPR r for
// lanes 0-15 and row r+8 for lanes 16-31, so lanes 0-15 take X,Y,Z from D1
// (pixels base+0..15) and lanes 16-31 take them from D2 (pixels base+16..31):
// every lane ends up with its own pixel's coordinates, no shuffles needed.

typedef float v2f __attribute__((ext_vector_type(2)));
typedef float v8f __attribute__((ext_vector_type(8)));

#define H_IMG 1024
#define W_IMG 1024

static __device__ __forceinline__ int iclamp(int v, int lo, int hi) {
    return v < lo ? lo : (v > hi ? hi : v);
}

__global__ void __launch_bounds__(256)
homography_warp_wmma_kernel(const float* __restrict__ img,
                            const float* __restrict__ homo,
                            float* __restrict__ out) {
    const int tid  = threadIdx.x;
    const int p    = blockIdx.x * 256 + tid;          // global pixel id
    const int hw   = p & 0xFFFFF;                     // h*1024 + w
    const int h    = hw >> 10;
    const int w    = hw & 1023;
    const int lane = tid & 31;

    // Batch index is block-uniform (2^20 % 256 == 0): force scalar expression
    // so the homography coefficients become s_loads.
    const int n = (blockIdx.x * 256) >> 20;
    const float* hm = homo + n * 9;
    const float h00 = hm[0], h01 = hm[1], h02 = hm[2];
    const float h10 = hm[3], h11 = hm[4], h12 = hm[5];
    const float h20 = hm[6], h21 = hm[7], h22 = hm[8];

    // ---- Build A (16x4 f32): rows 0-2 = homography, rows 8-10 duplicated ----
    // Layout: lanes 0-15 -> V0=A[M][0], V1=A[M][1]; lanes 16-31 -> V0=A[M][2], V1=A[M][3].
    const int  m  = lane & 7;                 // rows 8-10 mirror rows 0-2
    const bool r0 = (m == 0), r1 = (m == 1), r2 = (m == 2);
    const bool lo = (lane < 16);
    const float col0 = r0 ? h00 : (r1 ? h10 : (r2 ? h20 : 0.0f));
    const float col1 = r0 ? h01 : (r1 ? h11 : (r2 ? h21 : 0.0f));
    const float col2 = r0 ? h02 : (r1 ? h12 : (r2 ? h22 : 0.0f));
    v2f a;
    a[0] = lo ? col0 : col2;
    a[1] = lo ? col1 : 0.0f;

    // ---- Build B (4x16 f32) for the two 16-pixel halves of this wave ----
    // Layout: lanes 0-15 -> V0=B[0][n]=gx, V1=B[1][n]=gy;
    //         lanes 16-31 -> V0=B[2][n]=1, V1=B[3][n]=0.
    const float step = 2.0f / 1023.0f;                 // linspace(-1,1,1024)
    const int   wbase = w & ~31;                       // wave-uniform (W%32==0)
    const float gy  = fmaf((float)h, step, -1.0f);
    const float gx1 = fmaf((float)(wbase + (lane & 15)), step, -1.0f);
    const float gx2 = fmaf((float)(wbase + 16 + (lane & 15)), step, -1.0f);
    v2f b1, b2;
    b1[0] = lo ? gx1 : 1.0f;  b1[1] = lo ? gy : 0.0f;
    b2[0] = lo ? gx2 : 1.0f;  b2[1] = lo ? gy : 0.0f;

    v8f c0 = {};
    // 8 args: (neg_a, A, neg_b, B, c_mod, C, reuse_a, reuse_b)
    v8f d1 = __builtin_amdgcn_wmma_f32_16x16x4_f32(false, a, false, b1,
                                                   (short)0, c0, false, false);
    v8f d2 = __builtin_amdgcn_wmma_f32_16x16x4_f32(false, a, false, b2,
                                                   (short)0, c0, false, false);

    // Every lane's own pixel coordinates:
    const float X = lo ? d1[0] : d2[0];
    const float Y = lo ? d1[1] : d2[1];
    const float Z = lo ? d1[2] : d2[2];

    // ---- Projective divide + map to source pixel coords (align_corners) ----
    const float invz = 1.0f / Z;
    float sx = (fmaf(X, invz, 1.0f)) * (0.5f * (float)(W_IMG - 1));
    float sy = (fmaf(Y, invz, 1.0f)) * (0.5f * (float)(H_IMG - 1));
    // Clamp to keep int casts defined for degenerate Z; preserves OOB masking.
    sx = fminf(fmaxf(sx, -8.0f), (float)(W_IMG + 8));
    sy = fminf(fmaxf(sy, -8.0f), (float)(H_IMG + 8));

    const float fx0 = floorf(sx), fy0 = floorf(sy);
    const float fx = sx - fx0,    fy = sy - fy0;
    const int ix0 = (int)fx0, iy0 = (int)fy0;
    const int ix1 = ix0 + 1,  iy1 = iy0 + 1;

    const bool vx0 = (unsigned)ix0 < (unsigned)W_IMG;
    const bool vx1 = (unsigned)ix1 < (unsigned)W_IMG;
    const bool vy0 = (unsigned)iy0 < (unsigned)H_IMG;
    const bool vy1 = (unsigned)iy1 < (unsigned)H_IMG;
    const float m00 = (vx0 & vy0) ? 1.0f : 0.0f;
    const float m10 = (vx1 & vy0) ? 1.0f : 0.0f;
    const float m01 = (vx0 & vy1) ? 1.0f : 0.0f;
    const float m11 = (vx1 & vy1) ? 1.0f : 0.0f;

    const int cx0 = iclamp(ix0, 0, W_IMG - 1), cx1 = iclamp(ix1, 0, W_IMG - 1);
    const int cy0 = iclamp(iy0, 0, H_IMG - 1), cy1 = iclamp(iy1, 0, H_IMG - 1);
    const int l00 = cy0 * W_IMG + cx0, l10 = cy0 * W_IMG + cx1;
    const int l01 = cy1 * W_IMG + cx0, l11 = cy1 * W_IMG + cx1;

    const float wx1 = fx, wx0 = 1.0f - fx;
    const float wy1 = fy, wy0 = 1.0f - fy;
    const float w00 = wx0 * wy0, w10 = wx1 * wy0;
    const float w01 = wx0 * wy1, w11 = wx1 * wy1;

    const size_t nbase = ((size_t)n * 3) << 20;        // n*3*H*W
#pragma unroll
    for (int c = 0; c < 3; ++c) {
        const size_t plane = nbase + ((size_t)c << 20);
        const float* pp = img + plane;
        const float v00 = pp[l00] * m00;
        const float v10 = pp[l10] * m10;
        const float v01 = pp[l01] * m01;
        const float v11 = pp[l11] * m11;
        float r = w00 * v00;
        r = fmaf(w10, v10, r);
        r = fmaf(w01, v01, r);
        r = fmaf(w11, v11, r);
        out[plane + (size_t)hw] = r;
    }
}

extern "C" void kernel_launch(void* const* d_in, const int* in_sizes, int n_in,
                              void* d_out, int out_size, void* d_ws, size_t ws_size,
                              hipStream_t stream) {
    (void)in_sizes; (void)n_in; (void)out_size; (void)d_ws; (void)ws_size;
    const float* img  = (const float*)d_in[0];   // [16,3,1024,1024]
    const float* homo = (const float*)d_in[1];   // [16,3,3]
    float* out = (float*)d_out;                  // [16,3,1024,1024]

    const int total_pixels = 16 * H_IMG * W_IMG;     // 16,777,216
    const int block = 256;
    const int grid = total_pixels / block;           // 65,536 (exact, EXEC all-1s)
    homography_warp_wmma_kernel<<<grid, block, 0, stream>>>(img, homo, out);
}